// FocalLoss_54932631716288
// MI455X (gfx1250) — compile-verified
//
#include <hip/hip_runtime.h>
#include <math.h>

#define B_ 8
#define A_ 120000
#define C_ 9
#define M_ 64

__global__ __launch_bounds__(64) void fl_zero_ws(float* __restrict__ ws) {
    int i = threadIdx.x;
    if (i < 3 * B_) ws[i] = 0.0f;
}

__global__ __launch_bounds__(256) void fl_main(
    const float* __restrict__ cls,   // (B, A, 9)
    const float* __restrict__ reg,   // (B, A, 5)
    const float* __restrict__ anc,   // (1, A, 5)
    const float* __restrict__ ann,   // (B, M, 6)
    float* __restrict__ acc)         // (B, 3): cls_sum, reg_sum, num_pos
{
    __shared__ __align__(16) float s_raw[M_ * 6];
    __shared__ float4 s_box[M_];     // x1, y1, x2, y2
    __shared__ float4 s_aux[M_];     // area, sin(ang), cos(ang), valid
    __shared__ float  s_cls[M_];
    __shared__ float  s_acc[3];

    const int tid = threadIdx.x;
    const int b   = blockIdx.y;
    const int a   = blockIdx.x * 256 + tid;

    // ---- async stage: 64 annotations x 6 floats = 1536 B = 96 lanes x 16 B ----
    if (tid < (M_ * 6 * 4) / 16) {
        unsigned lds_off = ((unsigned)(size_t)&s_raw[0]) + (unsigned)tid * 16u;
        unsigned long long gaddr =
            (unsigned long long)(size_t)(ann + (size_t)b * M_ * 6 + (size_t)tid * 4);
        asm volatile("global_load_async_to_lds_b128 %0, %1, off"
                     :: "v"(lds_off), "v"(gaddr) : "memory");
    }
    asm volatile("s_wait_asynccnt 0" ::: "memory");
    __syncthreads();

    // ---- derive per-annotation quantities ----
    if (tid < M_) {
        float x1 = s_raw[tid * 6 + 0], y1 = s_raw[tid * 6 + 1];
        float x2 = s_raw[tid * 6 + 2], y2 = s_raw[tid * 6 + 3];
        float cv = s_raw[tid * 6 + 4], fl = s_raw[tid * 6 + 5];
        float sv, cc;
        __sincosf(cv, &sv, &cc);     // annotation "angle" = column 4 (class value)
        s_box[tid] = make_float4(x1, y1, x2, y2);
        s_aux[tid] = make_float4((x2 - x1) * (y2 - y1), sv, cc,
                                 (fl != -1.0f) ? 1.0f : 0.0f);
        s_cls[tid] = cv;
    }
    if (tid < 3) s_acc[tid] = 0.0f;
    __syncthreads();

    float cls_sum = 0.0f, reg_sum = 0.0f, posf = 0.0f;

    if (a < A_) {
        const float* ap = anc + (size_t)a * 5;
        float ax1 = ap[0], ay1 = ap[1], ax2 = ap[2], ay2 = ap[3], aang = ap[4];
        const float* pc = cls + ((size_t)b * A_ + (size_t)a) * C_;
        const float* pr = reg + ((size_t)b * A_ + (size_t)a) * 5;
        __builtin_prefetch(pc, 0, 3);   // global_prefetch_b8: hide cls/reg latency
        __builtin_prefetch(pr, 0, 3);   // behind the 64-iteration IoU loop

        float aw = ax2 - ax1, ah = ay2 - ay1, area_a = aw * ah;
        float sa, ca;
        __sincosf(aang, &sa, &ca);

        // ariou = iou * cos(|d| - pi/2) = iou * sin(|d|) = iou * |sin d|, |d| < pi.
        // Track max of num/den without dividing: num*bden > bnum*den (den > 0).
        float bnum = -3.0e38f, bden = 1.0f;
        int   barg = 0;
        #pragma unroll 4
        for (int m = 0; m < M_; ++m) {
            float4 bx = s_box[m];            // ds_load_b128 broadcast
            float4 bu = s_aux[m];            // ds_load_b128 broadcast
            float iw = fminf(ax2, bx.z) - fmaxf(ax1, bx.x);
            iw = fmaxf(iw, 0.0f);
            float ih = fminf(ay2, bx.w) - fmaxf(ay1, bx.y);
            ih = fmaxf(ih, 0.0f);
            float inter = iw * ih;
            float den   = fmaxf(area_a + bu.x - inter, 1e-8f);
            float sd    = fabsf(sa * bu.z - ca * bu.y);
            float num   = (bu.w != 0.0f) ? inter * sd : -1.0e9f;
            bool take = num * bden > bnum * den;   // strict '>' == first argmax
            bnum = take ? num : bnum;
            bden = take ? den : bden;
            barg = take ? m   : barg;
        }
        float best = bnum / bden;                  // single divide per anchor

        bool pos = (best >= 0.5f);
        bool neg = (best < 0.4f);
        posf = pos ? 1.0f : 0.0f;

        if (pos | neg) {
            int cidx = pos ? (int)s_cls[barg] : -1;
            #pragma unroll
            for (int c = 0; c < C_; ++c) {
                float p = pc[c];
                p = fminf(fmaxf(p, 1e-4f), 1.0f - 1e-4f);
                if (c == cidx)
                    cls_sum += 0.25f * (1.0f - p) * (1.0f - p) * (-__logf(p));
                else
                    cls_sum += 0.75f * p * p * (-__logf(1.0f - p));
            }
        }

        if (pos) {
            float acx = ax1 + 0.5f * aw, acy = ay1 + 0.5f * ah;
            float4 g = s_box[barg];
            float gwr = g.z - g.x, ghr = g.w - g.y;
            float gcx = g.x + 0.5f * gwr, gcy = g.y + 0.5f * ghr;
            float gw = fmaxf(gwr, 1.0f), gh = fmaxf(ghr, 1.0f);
            float t[5];
            t[0] = (gcx - acx) / aw * 10.0f;
            t[1] = (gcy - acy) / ah * 10.0f;
            t[2] = __logf(gw / aw) * 5.0f;
            t[3] = __logf(gh / ah) * 5.0f;
            t[4] = __tanf(s_cls[barg] - aang);
            #pragma unroll
            for (int j = 0; j < 5; ++j) {
                float d = fabsf(t[j] - pr[j]);
                reg_sum += (d <= (1.0f / 9.0f)) ? 4.5f * d * d : d - (0.5f / 9.0f);
            }
        }
    }

    // ---- wave32 tree reduction, then LDS atomics, then 3 global atomics/block ----
    #pragma unroll
    for (int off = 16; off > 0; off >>= 1) {
        cls_sum += __shfl_down(cls_sum, off, 32);
        reg_sum += __shfl_down(reg_sum, off, 32);
        posf    += __shfl_down(posf,    off, 32);
    }
    if ((tid & 31) == 0) {
        atomicAdd(&s_acc[0], cls_sum);
        atomicAdd(&s_acc[1], reg_sum);
        atomicAdd(&s_acc[2], posf);
    }
    __syncthreads();
    if (tid == 0) {
        atomicAdd(&acc[b * 3 + 0], s_acc[0]);
        atomicAdd(&acc[b * 3 + 1], s_acc[1]);
        atomicAdd(&acc[b * 3 + 2], s_acc[2]);
    }
}

// One wave32: lane b < B_ owns image b; rolled flag scan; shuffle-reduce the
// batch means. Keeps the kernel tiny (the old single-thread version was fully
// unrolled into ~2.5k scalar instructions).
__global__ __launch_bounds__(32) void fl_finalize(
    const float* __restrict__ ann, const float* __restrict__ acc,
    float* __restrict__ out)
{
    const int lane = threadIdx.x;
    float cl = 0.0f, rl = 0.0f;
    if (lane < B_) {
        bool has = false;
        #pragma clang loop unroll(disable)
        for (int m = 0; m < M_; ++m)
            has |= (ann[((size_t)lane * M_ + m) * 6 + 5] != -1.0f);
        float cs = acc[lane * 3 + 0];
        float rs = acc[lane * 3 + 1];
        float np = acc[lane * 3 + 2];
        cl = has ? cs / fmaxf(np, 1.0f) : 0.0f;
        rl = (has && np > 0.0f) ? rs / fmaxf(np * 5.0f, 1.0f) : 0.0f;
    }
    #pragma unroll
    for (int off = 16; off > 0; off >>= 1) {
        cl += __shfl_down(cl, off, 32);
        rl += __shfl_down(rl, off, 32);
    }
    if (lane == 0) {
        out[0] = cl * (1.0f / (float)B_);
        out[1] = rl * (1.0f / (float)B_);
    }
}

extern "C" void kernel_launch(void* const* d_in, const int* in_sizes, int n_in,
                              void* d_out, int out_size, void* d_ws, size_t ws_size,
                              hipStream_t stream) {
    const float* cls = (const float*)d_in[0];  // (B, A, 9)
    const float* reg = (const float*)d_in[1];  // (B, A, 5)
    const float* anc = (const float*)d_in[2];  // (1, A, 5)
    const float* ann = (const float*)d_in[3];  // (B, M, 6)
    float* acc = (float*)d_ws;                 // 24 floats used
    float* out = (float*)d_out;                // 2 floats

    fl_zero_ws<<<1, 64, 0, stream>>>(acc);
    dim3 grid((A_ + 255) / 256, B_);
    fl_main<<<grid, 256, 0, stream>>>(cls, reg, anc, ann, acc);
    fl_finalize<<<1, 32, 0, stream>>>(ann, acc, out);
}